// HungarianMatcher_70205535421152
// MI455X (gfx1250) — compile-verified
//
#include <hip/hip_runtime.h>
#include <stdint.h>

#define NUM_TGT 5
#define BLOCK   256
#define NCAND   3125   // 5^5
#define MAX_NQ  2048

typedef unsigned int u32x4 __attribute__((ext_vector_type(4)));
typedef int          i32x8 __attribute__((ext_vector_type(8)));
typedef int          i32x4 __attribute__((ext_vector_type(4)));

__device__ __forceinline__ uint32_t lds_addr_of(const void* p) {
  // flat pointer to __shared__ carries the LDS byte offset in its low 32 bits
  return (uint32_t)(uintptr_t)p;
}

__device__ __forceinline__ unsigned long long wave_min_u64(unsigned long long k) {
  #pragma unroll
  for (int off = 16; off > 0; off >>= 1) {
    unsigned long long o = __shfl_xor(k, off, 32);
    if (o < k) k = o;
  }
  return k;
}

__global__ __launch_bounds__(BLOCK)
void hm_lap5_kernel(const float* __restrict__ pred_regs,     // [bs, nq, 3]
                    const float* __restrict__ tgt_regression,// [bs*5, 3]
                    int* __restrict__ out_rows,              // [bs, 5]
                    int* __restrict__ out_cols,              // [bs, 5]
                    int nq) {
  const int b    = blockIdx.x;
  const int tid  = threadIdx.x;
  const int lane = tid & 31;
  const int wid  = tid >> 5;

  __shared__ __align__(16) float s_regs[MAX_NQ * 3];   // 24 KB staged query regs
  __shared__ __align__(16) float s_tgt[NUM_TGT * 3];   // staged target regs
  __shared__ float s_selv[NUM_TGT][NUM_TGT];           // top-5 costs per target
  __shared__ int   s_seli[NUM_TGT][NUM_TGT];           // top-5 query idx per target
  __shared__ unsigned long long s_red[BLOCK / 32];

  const float INF = 3.0e38f;
  const uint32_t nelem = (uint32_t)nq * 3u;            // f32 elements to stage

  // ---------- Phase 0a: wave 1 async-copies the 5 target rows (ASYNCcnt path) ----------
  if (wid == 1) {
    if (lane < NUM_TGT * 3) {
      const unsigned long long gb =
          (unsigned long long)(uintptr_t)(tgt_regression + (size_t)b * NUM_TGT * 3);
      uint32_t la = lds_addr_of(&s_tgt[0]) + (uint32_t)lane * 4u;
      uint32_t go = (uint32_t)lane * 4u;
      asm volatile("global_load_async_to_lds_b32 %0, %1, %2"
                   :: "v"(la), "v"(go), "s"(gb) : "memory");
    }
    asm volatile("s_wait_asynccnt 0" ::: "memory");
  }

  // ---------- Phase 0b: stage pred_regs[b] (24 KB) into LDS ----------
#if __has_builtin(__builtin_amdgcn_tensor_load_to_lds)
  // Single TDM DMA issued by wave 0 (TENSORcnt path).
  if (wid == 0) {
    const unsigned long long ga =
        (unsigned long long)(uintptr_t)(pred_regs + (size_t)b * nq * 3);
    const uint32_t lds0 = lds_addr_of(&s_regs[0]);

    u32x4 g0;
    g0.x = 1u;                                        // count=1, user mode
    g0.y = lds0;                                      // lds_addr
    g0.z = (uint32_t)(ga & 0xffffffffu);              // global_addr[31:0]
    g0.w = ((uint32_t)(ga >> 32) & 0x01ffffffu)       // global_addr[56:32]
           | 0x80000000u;                             // type=2 ("image")

    i32x8 g1;
    g1[0] = 0x00020000;                               // data_size=2 (4B), mask=0
    g1[1] = (int)((nelem & 0xffffu) << 16);           // tensor_dim0[15:0]
    g1[2] = (int)((nelem >> 16) | (1u << 16));        // tensor_dim0[31:16], tensor_dim1=1
    g1[3] = (int)((nelem & 0xffffu) << 16);           // tile_dim0 = nelem (fits 16b)
    g1[4] = 0;                                        // tile_dim1=0, tile_dim2=0
    g1[5] = (int)nelem;                               // tensor_dim0_stride[31:0]
    g1[6] = 0;
    g1[7] = 0;

    i32x4 gz = {0, 0, 0, 0};
#if __clang_major__ >= 23
    i32x8 gz8 = {0, 0, 0, 0, 0, 0, 0, 0};
    __builtin_amdgcn_tensor_load_to_lds(g0, g1, gz, gz, gz8, 0);
#else
    __builtin_amdgcn_tensor_load_to_lds(g0, g1, gz, gz, 0);
#endif
    __builtin_amdgcn_s_wait_tensorcnt((short)0);
  }
#else
  // Fallback: per-lane async b128 copies.
  {
    const unsigned long long gbase =
        (unsigned long long)(uintptr_t)(pred_regs + (size_t)b * nq * 3);
    const uint32_t bytes = nelem * 4u;
    const uint32_t lds0  = lds_addr_of(&s_regs[0]);
    for (uint32_t off = (uint32_t)tid * 16u; off < bytes; off += BLOCK * 16u) {
      uint32_t la = lds0 + off;
      asm volatile("global_load_async_to_lds_b128 %0, %1, %2"
                   :: "v"(la), "v"(off), "s"(gbase) : "memory");
    }
    asm volatile("s_wait_asynccnt 0" ::: "memory");
  }
#endif
  __syncthreads();

  // ---------- Phase 1: wave t finds the 5 cheapest queries for target t ----------
  if (wid < NUM_TGT) {
    const int t = wid;
    const float tx = s_tgt[t * 3 + 0];
    const float ty = s_tgt[t * 3 + 1];
    const float tz = s_tgt[t * 3 + 2];

    float bv[5] = {INF, INF, INF, INF, INF};
    int   bi[5] = {0, 0, 0, 0, 0};

    for (int q = lane; q < nq; q += 32) {
      const float dx = fabsf(s_regs[q * 3 + 0] - tx);
      const float dy = fabsf(s_regs[q * 3 + 1] - ty);
      const float dz = fabsf(s_regs[q * 3 + 2] - tz);
      const float c  = dx + dy + dz + 1.0f;   // + COST_CLASS
      if (c < bv[4]) {
        int pos = 4;
        while (pos > 0 && c < bv[pos - 1]) {
          bv[pos] = bv[pos - 1]; bi[pos] = bi[pos - 1]; --pos;
        }
        bv[pos] = c; bi[pos] = q;
      }
    }

    // cross-lane merge: 5 rounds of wave-argmin with pop (keys unique: idx in low bits)
    #pragma unroll
    for (int k = 0; k < NUM_TGT; ++k) {
      unsigned long long key =
          (((unsigned long long)__float_as_uint(bv[0])) << 32) | (unsigned)bi[0];
      unsigned long long rk = wave_min_u64(key);
      if (key == rk) {   // this lane supplied the winner -> pop its list head
        bv[0] = bv[1]; bi[0] = bi[1];
        bv[1] = bv[2]; bi[1] = bi[2];
        bv[2] = bv[3]; bi[2] = bi[3];
        bv[3] = bv[4]; bi[3] = bi[4];
        bv[4] = INF;   bi[4] = 0;
      }
      if (lane == 0) {
        s_selv[t][k] = __uint_as_float((uint32_t)(rk >> 32));
        s_seli[t][k] = (int)(rk & 0xffffffffu);
      }
    }
  }
  __syncthreads();

  // ---------- Phase 2: enumerate 5^5 candidate tuples, masked argmin ----------
  float vv[25]; int qq[25];
  #pragma unroll
  for (int i = 0; i < 25; ++i) {
    vv[i] = s_selv[i / 5][i % 5];
    qq[i] = s_seli[i / 5][i % 5];
  }

  unsigned long long bestkey = ~0ull;
  for (int c = tid; c < NCAND; c += BLOCK) {
    int d = c;
    const int d0 = d % 5; d /= 5;
    const int d1 = d % 5; d /= 5;
    const int d2 = d % 5; d /= 5;
    const int d3 = d % 5; d /= 5;
    const int d4 = d;
    const int q0 = qq[0 + d0], q1 = qq[5 + d1], q2 = qq[10 + d2],
              q3 = qq[15 + d3], q4 = qq[20 + d4];
    const float s = vv[0 + d0] + vv[5 + d1] + vv[10 + d2] + vv[15 + d3] + vv[20 + d4];
    const bool valid = (q0 != q1) & (q0 != q2) & (q0 != q3) & (q0 != q4) &
                       (q1 != q2) & (q1 != q3) & (q1 != q4) &
                       (q2 != q3) & (q2 != q4) & (q3 != q4);
    const unsigned long long key =
        valid ? ((((unsigned long long)__float_as_uint(s)) << 32) | (unsigned)c)
              : ~0ull;
    if (key < bestkey) bestkey = key;
  }
  bestkey = wave_min_u64(bestkey);
  if (lane == 0) s_red[wid] = bestkey;
  __syncthreads();

  // ---------- Phase 3: decode winner, sort, write ----------
  if (tid == 0) {
    unsigned long long bk = s_red[0];
    #pragma unroll
    for (int w = 1; w < BLOCK / 32; ++w)
      if (s_red[w] < bk) bk = s_red[w];

    int c = (int)(bk & 0xffffffffu);
    int rows[NUM_TGT], cols[NUM_TGT];
    #pragma unroll
    for (int t = 0; t < NUM_TGT; ++t) {
      rows[t] = s_seli[t][c % 5];
      cols[t] = t;
      c /= 5;
    }
    // sort (row, target) pairs by row ascending -> row_ind, col_ind = argsort(rows)
    #pragma unroll
    for (int i = 0; i < NUM_TGT - 1; ++i)
      #pragma unroll
      for (int j = 0; j < NUM_TGT - 1 - i; ++j)
        if (rows[j] > rows[j + 1]) {
          int tr = rows[j]; rows[j] = rows[j + 1]; rows[j + 1] = tr;
          int tc = cols[j]; cols[j] = cols[j + 1]; cols[j + 1] = tc;
        }
    #pragma unroll
    for (int t = 0; t < NUM_TGT; ++t) {
      out_rows[b * NUM_TGT + t] = rows[t];
      out_cols[b * NUM_TGT + t] = cols[t];
    }
  }
}

extern "C" void kernel_launch(void* const* d_in, const int* in_sizes, int n_in,
                              void* d_out, int out_size, void* d_ws, size_t ws_size,
                              hipStream_t stream) {
  // setup_inputs order: 0 pred_logits (unused), 1 pred_regs, 2 labels (unused), 3 tgt_regression
  const float* pred_regs = (const float*)d_in[1];
  const float* tgt       = (const float*)d_in[3];

  const int T  = in_sizes[2];            // bs * NUM_TGT
  const int bs = T / NUM_TGT;            // 32
  int nq = in_sizes[1] / (bs * 3);       // 2048
  if (nq > MAX_NQ) nq = MAX_NQ;

  int* out_rows = (int*)d_out;           // first bs*5 ints
  int* out_cols = out_rows + bs * NUM_TGT;

  hm_lap5_kernel<<<bs, BLOCK, 0, stream>>>(pred_regs, tgt, out_rows, out_cols, nq);
}